// AdditiveAttention_38208029065916
// MI455X (gfx1250) — compile-verified
//
#include <hip/hip_runtime.h>

typedef __attribute__((ext_vector_type(2))) float v2f;
typedef __attribute__((ext_vector_type(8))) float v8f;

#define B_    32
#define TQ    64
#define TK    256
#define H_    64
#define QCH   16      // query rows per block
#define KHALF 128     // k rows per LDS pass
#define KS    65      // kproj LDS row stride (bank-conflict pad)
#define QS    65      // qproj LDS row stride
#define SS    260     // scores LDS row stride (pad=4 -> bank stride 4 for WMMA A reads)

#define NEGPAD (-4294967295.0f)   // -2^32 + 1
#define LOG2E  1.4426950408889634f

// Prefer the native v_tanh_f32 transcendental (CDNA5 TRANS op) if clang
// exposes it for gfx1250; otherwise 1x v_exp_f32 + 1x v_rcp_f32.
__device__ __forceinline__ float fast_tanh(float x) {
#if __has_builtin(__builtin_amdgcn_tanhf)
  return __builtin_amdgcn_tanhf(x);
#else
  float e = __builtin_amdgcn_exp2f(x * (2.0f * LOG2E));
  float r = __builtin_amdgcn_rcpf(e + 1.0f);
  return __builtin_fmaf(-2.0f, r, 1.0f);
#endif
}

__global__ __launch_bounds__(256)
void AdditiveAttention_kernel(const float* __restrict__ queries,
                              const float* __restrict__ keys,
                              const int*   __restrict__ seq_len,
                              const float* __restrict__ Wq,
                              const float* __restrict__ Wk,
                              const float* __restrict__ vvec,
                              const float* __restrict__ bias,
                              float* __restrict__ out)
{
  __shared__ float kp[KHALF * KS];   // projected keys (one half of Tk)
  __shared__ float qpb[QCH * QS];    // projected queries + bias
  __shared__ float sc[QCH * SS];     // scores -> softmax weights
  __shared__ float vsh[H_];

  const int tid  = threadIdx.x;
  const int wave = tid >> 5;
  const int lane = tid & 31;
  const int r    = lane & 15;   // row/col index within 16-wide fragment
  const int hi   = lane >> 4;   // half-wave select (K 0-1 vs 2-3; M base 0 vs 8)

  const int b  = blockIdx.x >> 2;
  const int q0 = (blockIdx.x & 3) * QCH;

  const float* qbase = queries + (size_t)b * TQ * H_;
  const float* kbase = keys    + (size_t)b * TK * H_;
  const int sl = seq_len[b];

  // prefetch this batch's key rows (global_prefetch_b8); L2-hot thereafter
  __builtin_prefetch(kbase + (size_t)tid * H_, 0, 0);

  if (tid < H_) vsh[tid] = vvec[tid];

  // ---- phase 1a: q projection + bias -> qpb[16][64] (waves 0-3, WMMA f32) ----
  if (wave < 4) {
    const int n0 = wave * 16;
    v8f acc = {};
    for (int kt = 0; kt < 16; ++kt) {
      const int k0 = kt * 4 + 2 * hi;
      const float* ap = qbase + (q0 + r) * H_ + k0;
      v2f a;  a.x = ap[0];               a.y = ap[1];
      v2f bb; bb.x = Wq[k0 * H_ + n0 + r]; bb.y = Wq[(k0 + 1) * H_ + n0 + r];
      acc = __builtin_amdgcn_wmma_f32_16x16x4_f32(false, a, false, bb,
                                                  (short)0, acc, false, false);
    }
    const int col = n0 + r;
    const float bc = bias[col];
    #pragma unroll
    for (int i = 0; i < 8; ++i)
      qpb[(8 * hi + i) * QS + col] = acc[i] + bc;
  }
  __syncthreads();

  // ---- phases 1b+2: k projection (WMMA) then tanh-dot scores, in 2 halves ----
  for (int half = 0; half < 2; ++half) {
    const int kb = half * KHALF;

    // k projection for rows [kb, kb+128): 8x4 tiles, 4 per wave
    #pragma unroll
    for (int t = 0; t < 4; ++t) {
      const int tile = wave + 8 * t;        // 0..31
      const int m0 = (tile >> 2) * 16;      // 0..112
      const int n0 = (tile & 3) * 16;
      v8f acc = {};
      for (int kt = 0; kt < 16; ++kt) {
        const int k0 = kt * 4 + 2 * hi;
        const float* ap = kbase + (kb + m0 + r) * H_ + k0;
        v2f a;  a.x = ap[0];               a.y = ap[1];
        v2f bb; bb.x = Wk[k0 * H_ + n0 + r]; bb.y = Wk[(k0 + 1) * H_ + n0 + r];
        acc = __builtin_amdgcn_wmma_f32_16x16x4_f32(false, a, false, bb,
                                                    (short)0, acc, false, false);
      }
      #pragma unroll
      for (int i = 0; i < 8; ++i)
        kp[(m0 + 8 * hi + i) * KS + n0 + r] = acc[i];
    }
    __syncthreads();

    // scores: s[q][kk] = sum_h v[h]*tanh(qpb[q][h] + kp[kk][h])
    // thread -> q = tid>>4 (broadcast qpb), kk strided by 16 (conflict-free kp)
    const int q  = tid >> 4;
    const int c0 = tid & 15;
    #pragma unroll
    for (int p = 0; p < 8; ++p) {
      const int kkl = c0 + 16 * p;
      float acc = 0.0f;
      #pragma unroll 16
      for (int h = 0; h < H_; ++h) {
        float x = qpb[q * QS + h] + kp[kkl * KS + h];
        acc = __builtin_fmaf(vsh[h], fast_tanh(x), acc);
      }
      sc[q * SS + kb + kkl] = acc;
    }
    __syncthreads();
  }

  // ---- phase 3: masked softmax over Tk (2 rows per wave) ----
  #pragma unroll
  for (int rr = 0; rr < 2; ++rr) {
    const int row = wave * 2 + rr;
    float vals[8];
    float m = NEGPAD;
    #pragma unroll
    for (int j = 0; j < 8; ++j) {
      const int kk = lane + 32 * j;
      float x = sc[row * SS + kk];
      x = (kk < sl) ? x : NEGPAD;
      vals[j] = x;
      m = fmaxf(m, x);
    }
    #pragma unroll
    for (int off = 16; off > 0; off >>= 1)
      m = fmaxf(m, __shfl_xor(m, off, 32));
    float sum = 0.0f;
    #pragma unroll
    for (int j = 0; j < 8; ++j) {
      float e = __builtin_amdgcn_exp2f((vals[j] - m) * LOG2E);
      vals[j] = e;
      sum += e;
    }
    #pragma unroll
    for (int off = 16; off > 0; off >>= 1)
      sum += __shfl_xor(sum, off, 32);
    const float inv = 1.0f / sum;
    #pragma unroll
    for (int j = 0; j < 8; ++j)
      sc[row * SS + lane + 32 * j] = vals[j] * inv;
  }
  __syncthreads();

  // ---- phase 4: out[16][64] = align[16][256] @ keys[256][64] (WMMA f32) ----
  if (wave < 4) {
    const int n0 = wave * 16;
    v8f acc = {};
    for (int kt = 0; kt < 64; ++kt) {
      const int k0 = kt * 4 + 2 * hi;
      v2f a;  a.x = sc[r * SS + k0];            a.y = sc[r * SS + k0 + 1];
      v2f bb; bb.x = kbase[k0 * H_ + n0 + r];   bb.y = kbase[(k0 + 1) * H_ + n0 + r];
      acc = __builtin_amdgcn_wmma_f32_16x16x4_f32(false, a, false, bb,
                                                  (short)0, acc, false, false);
    }
    #pragma unroll
    for (int i = 0; i < 8; ++i)
      out[((size_t)b * TQ + q0 + 8 * hi + i) * H_ + n0 + r] = acc[i];
  }
}

extern "C" void kernel_launch(void* const* d_in, const int* in_sizes, int n_in,
                              void* d_out, int out_size, void* d_ws, size_t ws_size,
                              hipStream_t stream) {
  const float* queries = (const float*)d_in[0];
  const float* keys    = (const float*)d_in[1];
  const int*   seq_len = (const int*)  d_in[2];
  const float* Wq      = (const float*)d_in[3];
  const float* Wk      = (const float*)d_in[4];
  const float* v       = (const float*)d_in[5];
  const float* b       = (const float*)d_in[6];
  float* out = (float*)d_out;

  dim3 grid(B_ * (TQ / QCH));   // 32 batches x 4 query chunks = 128 blocks
  dim3 block(256);              // 8 wave32
  hipLaunchKernelGGL(AdditiveAttention_kernel, grid, block, 0, stream,
                     queries, keys, seq_len, Wq, Wk, v, b, out);
}